// GNN_10969346474643
// MI455X (gfx1250) — compile-verified
//
#include <hip/hip_runtime.h>

#define NNODES 50000
#define NEDGES 800000
#define NEG_SLOPE 0.2f

typedef float v2f __attribute__((ext_vector_type(2)));
typedef float v8f __attribute__((ext_vector_type(8)));

static inline int ceil_div(int a, int b) { return (a + b - 1) / b; }

// ---------------------------------------------------------------------------
// CSR build (graph is identical for every layer/batch -> build once per launch)
// ---------------------------------------------------------------------------
__global__ void count_deg(const int* __restrict__ dst, int* __restrict__ deg) {
  int e = blockIdx.x * blockDim.x + threadIdx.x;
  if (e < NEDGES) atomicAdd(&deg[dst[e]], 1);
}

// single-block exclusive scan (Hillis-Steele over 1024-chunks + running carry)
__global__ void exscan_rowptr(const int* __restrict__ deg, int* __restrict__ rowptr,
                              int n) {
  __shared__ int smem[1024];
  __shared__ int carry;
  const int tid = threadIdx.x;
  if (tid == 0) carry = 0;
  __syncthreads();
  for (int base = 0; base < n; base += 1024) {
    int i = base + tid;
    int v = (i < n) ? deg[i] : 0;
    smem[tid] = v;
    __syncthreads();
    for (int off = 1; off < 1024; off <<= 1) {
      int t = (tid >= off) ? smem[tid - off] : 0;
      __syncthreads();
      smem[tid] += t;
      __syncthreads();
    }
    int incl = smem[tid];
    int c = carry;
    if (i < n) rowptr[i] = c + incl - v;
    __syncthreads();
    if (tid == 0) carry = c + smem[1023];
    __syncthreads();
  }
  if (tid == 0) rowptr[n] = carry;
}

__global__ void scatter_csr(const int* __restrict__ src, const int* __restrict__ dst,
                            const int* __restrict__ rowptr, int* __restrict__ fill,
                            int* __restrict__ csr_src) {
  int e = blockIdx.x * blockDim.x + threadIdx.x;
  if (e >= NEDGES) return;
  int d = dst[e];
  int pos = rowptr[d] + atomicAdd(&fill[d], 1);
  csr_src[pos] = src[e];
}

// ---------------------------------------------------------------------------
// Dense projection: C[M,Nc] = A[M,K] @ B[K,Nc] with V_WMMA_F32_16X16X4_F32.
// Block = 256 threads (8 waves) owns one 16-row strip of A, staged into LDS
// via gfx1250 async global->LDS copies; wave w computes the 16x16 tile at
// column w*16.  Fragment layouts per CDNA5 ISA 7.12.2 (fp32 16x4 / 4x16):
//   A: lanes 0-15 row=lane  v0=K+0 v1=K+1 ; lanes 16-31 same rows, K+2/K+3
//   B: lanes 0-15 col=lane  v0=K+0 v1=K+1 ; lanes 16-31 same cols, K+2/K+3
//   C: vgpr v -> (M=v, N=lane) / lanes 16-31 -> (M=v+8, N=lane-16)
// LDS row stride = K+4 floats so the 16-lane stride-K fragment reads map to
// distinct banks (l16*4 mod 64 all distinct).
// ---------------------------------------------------------------------------
__global__ void gemm_wmma_lds(const float* __restrict__ A,
                              const float* __restrict__ Bm,
                              float* __restrict__ C,
                              int K, int logK, int Nc, int KP) {
  __shared__ float As[16 * 132];  // max K=128 -> KP=132
  const int tid = threadIdx.x;
  const int lane = tid & 31;
  const int w = tid >> 5;
  const int tm = blockIdx.x << 4;  // 50000/16 = 3125 strips exactly

  // cooperative async copy A[tm:tm+16, 0:K] -> LDS (coalesced rows)
  const int elems = 16 << logK;
  const unsigned ldsBase = (unsigned)(size_t)&As[0];  // low 32b of flat = LDS offset
  const int cmask = (1 << logK) - 1;
  for (int i = tid; i < elems; i += 256) {
    int r = i >> logK, c = i & cmask;
    const float* gp = A + ((size_t)(tm + r) << logK) + c;
    unsigned loff = ldsBase + (unsigned)(r * KP + c) * 4u;
    asm volatile("global_load_async_to_lds_b32 %0, %1, off"
                 :: "v"(loff), "v"(gp) : "memory");
  }
  asm volatile("s_wait_asynccnt 0x0" ::: "memory");
  __syncthreads();

  const int tilesN = Nc >> 4;
  if (w < tilesN) {                       // wave-uniform branch, EXEC stays full
    const int tn = w << 4;
    const int half = lane >> 4;
    const int l16 = lane & 15;
    const float* arow = &As[l16 * KP + 2 * half];
    const float* bcol = Bm + (size_t)(2 * half) * Nc + (tn + l16);
    v8f acc = {};
    for (int k0 = 0; k0 < K; k0 += 4) {
      v2f av, bv;
      av.x = arow[k0];
      av.y = arow[k0 + 1];
      bv.x = bcol[(size_t)k0 * Nc];
      bv.y = bcol[(size_t)k0 * Nc + Nc];
      acc = __builtin_amdgcn_wmma_f32_16x16x4_f32(false, av, false, bv,
                                                  (short)0, acc, false, false);
    }
    float* crow = C + (size_t)(tm + 8 * half) * Nc + (tn + l16);
#pragma unroll
    for (int v = 0; v < 8; ++v) crow[(size_t)v * Nc] = acc[v];
  }
}

// ---------------------------------------------------------------------------
// el[n,h] = sum_d h[n,h,d]*al[h,d] ; er likewise. One wave32 per (node,head).
// ---------------------------------------------------------------------------
__global__ void attn_scores(const float* __restrict__ h,
                            const float* __restrict__ al,
                            const float* __restrict__ ar,
                            float* __restrict__ el, float* __restrict__ er,
                            int H, int D, int hshift) {
  const int lane = threadIdx.x & 31;
  const int wave = (int)((blockIdx.x * blockDim.x + threadIdx.x) >> 5);
  if (wave >= NNODES * H) return;
  const int n = wave >> hshift;
  const int hh = wave & (H - 1);
  const float* hp = h + ((size_t)n * H + hh) * D;
  const float* alp = al + (size_t)hh * D;
  const float* arp = ar + (size_t)hh * D;
  float sl = 0.f, sr = 0.f;
  for (int d = lane; d < D; d += 32) {
    float v = hp[d];
    sl += v * alp[d];
    sr += v * arp[d];
  }
#pragma unroll
  for (int off = 16; off > 0; off >>= 1) {
    sl += __shfl_down(sl, off, 32);
    sr += __shfl_down(sr, off, 32);
  }
  if (lane == 0) {
    el[((size_t)n << hshift) + hh] = sl;
    er[((size_t)n << hshift) + hh] = sr;
  }
}

__device__ __forceinline__ float leaky(float v) {
  return (v > 0.f) ? v : NEG_SLOPE * v;
}

// ---------------------------------------------------------------------------
// Fused per-node GAT tail: one wave32 per destination node.
//   pass 1: segment max of leaky(el[src]+er[dst]) per head
//   pass 2: segment sum of exp(e - m) per head
//   pass 3: out[d,f] = sum_j h[src_j,f] * exp(e_j-m)/s   (32-edge batches,
//           weights computed lane-parallel then __shfl-broadcast)
//   + bias + optional relu.  Zero atomics, zero intermediate edge buffers.
// ---------------------------------------------------------------------------
__global__ void gat_node_fused(const int* __restrict__ rowptr,
                               const int* __restrict__ csr_src,
                               const float* __restrict__ P,
                               const float* __restrict__ el,
                               const float* __restrict__ er,
                               const float* __restrict__ bias,
                               float* __restrict__ out,
                               int H, int hshift, int F, int act) {
  const int lane = threadIdx.x & 31;
  const int node = (int)((blockIdx.x * blockDim.x + threadIdx.x) >> 5);
  if (node >= NNODES) return;
  const int beg = rowptr[node], end = rowptr[node + 1];
  const int hb = F >> hshift;  // head boundary in flattened features
  const float er0 = er[(size_t)node << hshift];
  const float er1 = (H == 2) ? er[((size_t)node << hshift) + 1] : 0.f;

  // pass 1: max per head
  float m0 = -3.4e38f, m1 = -3.4e38f;
  for (int j = beg + lane; j < end; j += 32) {
    int s = csr_src[j];
    m0 = fmaxf(m0, leaky(el[(size_t)s << hshift] + er0));
    if (H == 2) m1 = fmaxf(m1, leaky(el[((size_t)s << hshift) + 1] + er1));
  }
#pragma unroll
  for (int off = 16; off > 0; off >>= 1) {
    m0 = fmaxf(m0, __shfl_xor(m0, off, 32));
    if (H == 2) m1 = fmaxf(m1, __shfl_xor(m1, off, 32));
  }
  // pass 2: sum of exp per head
  float s0 = 0.f, s1 = 0.f;
  for (int j = beg + lane; j < end; j += 32) {
    int s = csr_src[j];
    s0 += __expf(leaky(el[(size_t)s << hshift] + er0) - m0);
    if (H == 2) s1 += __expf(leaky(el[((size_t)s << hshift) + 1] + er1) - m1);
  }
#pragma unroll
  for (int off = 16; off > 0; off >>= 1) {
    s0 += __shfl_xor(s0, off, 32);
    if (H == 2) s1 += __shfl_xor(s1, off, 32);
  }
  const float inv0 = (s0 > 0.f) ? 1.f / s0 : 0.f;
  const float inv1 = (s1 > 0.f) ? 1.f / s1 : 0.f;

  // pass 3: weighted feature accumulation (coalesced rows of P)
  float acc[4] = {0.f, 0.f, 0.f, 0.f};
  for (int base = beg; base < end; base += 32) {
    int j = base + lane;
    int sIdx = 0;
    float w0 = 0.f, w1 = 0.f;
    if (j < end) {
      sIdx = csr_src[j];
      w0 = __expf(leaky(el[(size_t)sIdx << hshift] + er0) - m0) * inv0;
      w1 = (H == 2)
               ? __expf(leaky(el[((size_t)sIdx << hshift) + 1] + er1) - m1) * inv1
               : w0;
    }
    const int cnt = min(32, end - base);
    for (int t = 0; t < cnt; ++t) {
      int s = __shfl(sIdx, t, 32);
      float ww0 = __shfl(w0, t, 32);
      float ww1 = __shfl(w1, t, 32);
      const float* hp = P + (size_t)s * F;
#pragma unroll
      for (int c = 0; c < 4; ++c) {
        int f = lane + 32 * c;
        if (f < F) acc[c] += hp[f] * ((f < hb) ? ww0 : ww1);
      }
    }
  }
  // bias + activation + store
#pragma unroll
  for (int c = 0; c < 4; ++c) {
    int f = lane + 32 * c;
    if (f < F) {
      float v = acc[c] + bias[f];
      if (act) v = fmaxf(v, 0.f);
      out[(size_t)node * F + f] = v;
    }
  }
}

// ---------------------------------------------------------------------------
static void run_gat_layer(const float* Xin, const float* W, const float* al,
                          const float* ar, const float* bias,
                          const int* rowptr, const int* csr_src,
                          int K, int logK, int H, int D, int hshift, int act,
                          float* P, float* Xout, float* el, float* er,
                          hipStream_t stream) {
  const int F = H * D;
  gemm_wmma_lds<<<NNODES / 16, 256, 0, stream>>>(Xin, W, P, K, logK, F, K + 4);
  attn_scores<<<ceil_div(NNODES * H, 8), 256, 0, stream>>>(P, al, ar, el, er, H,
                                                           D, hshift);
  gat_node_fused<<<ceil_div(NNODES, 8), 256, 0, stream>>>(
      rowptr, csr_src, P, el, er, bias, Xout, H, hshift, F, act);
}

extern "C" void kernel_launch(void* const* d_in, const int* in_sizes, int n_in,
                              void* d_out, int out_size, void* d_ws, size_t ws_size,
                              hipStream_t stream) {
  const float* input = (const float*)d_in[0];  // [B, N, 64]
  const int* src = (const int*)d_in[1];
  const int* dst = (const int*)d_in[2];
  const float* W1 = (const float*)d_in[3];
  const float* al1 = (const float*)d_in[4];
  const float* ar1 = (const float*)d_in[5];
  const float* b1 = (const float*)d_in[6];
  const float* W2 = (const float*)d_in[7];
  const float* al2 = (const float*)d_in[8];
  const float* ar2 = (const float*)d_in[9];
  const float* b2 = (const float*)d_in[10];
  const float* W3 = (const float*)d_in[11];
  const float* al3 = (const float*)d_in[12];
  const float* ar3 = (const float*)d_in[13];
  const float* b3 = (const float*)d_in[14];
  float* out = (float*)d_out;

  // workspace carve-out (~56 MB)
  float* P = (float*)d_ws;                         // [N,128] projections
  float* X = P + (size_t)NNODES * 128;             // [N,128] activations
  float* el = X + (size_t)NNODES * 128;            // [N,2]
  float* er = el + (size_t)NNODES * 2;             // [N,2]
  int* deg = (int*)(er + (size_t)NNODES * 2);      // [N]
  int* rowptr = deg + NNODES;                      // [N+1]
  int* fill = rowptr + (NNODES + 1);               // [N]
  int* csr_src = fill + NNODES;                    // [E]

  // ---- CSR build: once per launch, reused by all 6 layer passes ----
  hipMemsetAsync(deg, 0, sizeof(int) * NNODES, stream);
  hipMemsetAsync(fill, 0, sizeof(int) * NNODES, stream);
  count_deg<<<ceil_div(NEDGES, 256), 256, 0, stream>>>(dst, deg);
  exscan_rowptr<<<1, 1024, 0, stream>>>(deg, rowptr, NNODES);
  scatter_csr<<<ceil_div(NEDGES, 256), 256, 0, stream>>>(src, dst, rowptr, fill,
                                                         csr_src);

  for (int b = 0; b < 2; ++b) {
    const float* x0 = input + (size_t)b * NNODES * 64;
    float* yb = out + (size_t)b * NNODES * 64;
    // layer 1: GATConv(64 -> 64, heads=2), flatten -> [N,128]
    run_gat_layer(x0, W1, al1, ar1, b1, rowptr, csr_src, 64, 6, 2, 64, 1, 0, P,
                  X, el, er, stream);
    // layer 2: GATConv(128 -> 128, heads=1), relu
    run_gat_layer(X, W2, al2, ar2, b2, rowptr, csr_src, 128, 7, 1, 128, 0, 1, P,
                  X, el, er, stream);
    // layer 3: GATConv(128 -> 64, heads=1) -> output slice
    run_gat_layer(X, W3, al3, ar3, b3, rowptr, csr_src, 128, 7, 1, 64, 0, 0, P,
                  yb, el, er, stream);
  }
}